// PhysicsInformedLoss_3667902070834
// MI455X (gfx1250) — compile-verified
//
#include <hip/hip_runtime.h>
#include <stdint.h>

// PhysicsInformedLoss for MI455X (gfx1250, wave32).
// Memory-bound reduction workload; hash-table replaces the reference's
// sort+searchsorted for the antisymmetry term. Deterministic two-stage
// reductions (no float atomics). WMMA f32 16x16x4 used (exactly) for the
// per-molecule conservation row-sums. Streaming arrays use non-temporal
// loads (th:TH_LOAD_NT) so the 256 MiB hash table + 67 MB gather target
// keep L2 (192 MB) residency during the probe pass.

typedef unsigned int       u32;
typedef unsigned long long u64;

#define NT        256      // threads per block (8 waves of 32)
#define NB_EDGE   2048     // blocks for edge pass
#define NB_NODE   32       // blocks for node pass
#define NB_CONS   8        // blocks for conservation pass (8*8 waves * 16 mol = 1024)

#define NN        32768
#define NE        16777216
#define NMOL      1024

#define HASH_EMPTY (~0ull)
#define MAX_PROBE  4096u

typedef __attribute__((ext_vector_type(2))) float v2f;
typedef __attribute__((ext_vector_type(8))) float v8f;
typedef __attribute__((ext_vector_type(4))) float f32x4;
typedef __attribute__((ext_vector_type(4))) int   i32x4;
typedef __attribute__((ext_vector_type(2))) unsigned long long u64x2;

static __device__ __forceinline__ u32 hmix(u32 k){
  k *= 0x9E3779B1u; k ^= k >> 15; k *= 0x85EBCA6Bu; k ^= k >> 13; return k;
}

// slot = (key << 32) | edge_index. For duplicate keys, atomicMax of the packed
// word selects the LAST (max) original index, matching the reference dict.
static __device__ __forceinline__ void hins(u64* T, u32 key, u32 e, u32 mask){
  u32 s = hmix(key) & mask;
  u64 pk = ((u64)key << 32) | (u64)e;
  for (u32 p = 0; p < MAX_PROBE; ++p){
    u64 cur = atomicCAS(&T[s], HASH_EMPTY, pk);
    if (cur == HASH_EMPTY) return;
    if ((u32)(cur >> 32) == key){ atomicMax(&T[s], pk); return; }
    s = (s + 1) & mask;
  }
}

static __device__ __forceinline__ int hfind(const u64* __restrict__ T, u32 key, u32 mask){
  u32 s = hmix(key) & mask;
  for (u32 p = 0; p < MAX_PROBE; ++p){
    u64 cur = T[s];                 // temporal: we want these L2-resident
    if (cur == HASH_EMPTY) return -1;
    if ((u32)(cur >> 32) == key) return (int)(u32)cur;
    s = (s + 1) & mask;
  }
  return -1;
}

static __device__ __forceinline__ float block_reduce_f(float v, float* sh){
  sh[threadIdx.x] = v; __syncthreads();
  #pragma unroll
  for (int s = NT/2; s > 0; s >>= 1){
    if ((int)threadIdx.x < s) sh[threadIdx.x] += sh[threadIdx.x + s];
    __syncthreads();
  }
  float r = sh[0]; __syncthreads(); return r;
}

static __device__ __forceinline__ u32 block_reduce_u(u32 v, u32* sh){
  sh[threadIdx.x] = v; __syncthreads();
  #pragma unroll
  for (int s = NT/2; s > 0; s >>= 1){
    if ((int)threadIdx.x < s) sh[threadIdx.x] += sh[threadIdx.x + s];
    __syncthreads();
  }
  u32 r = sh[0]; __syncthreads(); return r;
}

static __device__ __forceinline__ float sq4(f32x4 a, f32x4 b){
  float dx = a.x-b.x, dy = a.y-b.y, dz = a.z-b.z, dw = a.w-b.w;
  return dx*dx + dy*dy + dz*dz + dw*dw;
}
static __device__ __forceinline__ float relu2(float x){ float m = fmaxf(x, 0.f); return m*m; }

// ---------------------------------------------------------------- hash init
// NT b128 stores: the table is re-read via build atomics and plain probe loads (L2), no need to
// pollute near caches while filling 256 MiB.
__global__ void k_init_hash(u64* __restrict__ T, u32 cap){
  u64x2 e; e.x = HASH_EMPTY; e.y = HASH_EMPTY;
  u64x2* T2 = (u64x2*)T;
  const u32 n2 = cap >> 1;
  const u32 stride = gridDim.x * blockDim.x;
  for (u32 i = blockIdx.x*blockDim.x + threadIdx.x; i < n2; i += stride)
    __builtin_nontemporal_store(e, T2 + i);
}

// ---------------------------------------------------------------- hash build
__global__ void k_build_hash(const int* __restrict__ ei, u64* __restrict__ T, u32 mask){
  const i32x4* i4 = (const i32x4*)ei;
  const i32x4* j4 = (const i32x4*)(ei + NE);
  const u32 nq = NE/4;
  const u32 stride = gridDim.x * blockDim.x;
  for (u32 q = blockIdx.x*blockDim.x + threadIdx.x; q < nq; q += stride){
    i32x4 iv = __builtin_nontemporal_load(i4 + q);   // streamed once
    i32x4 jv = __builtin_nontemporal_load(j4 + q);
    u32 e = q * 4u;
    hins(T, (u32)iv.x*32768u + (u32)jv.x, e+0u, mask);
    hins(T, (u32)iv.y*32768u + (u32)jv.y, e+1u, mask);
    hins(T, (u32)iv.z*32768u + (u32)jv.z, e+2u, mask);
    hins(T, (u32)iv.w*32768u + (u32)jv.w, e+3u, mask);
  }
}

// ------------------------------------------------- fused kbo MSE + antisym
__global__ void k_edge(const float* __restrict__ kp, const float* __restrict__ kt,
                       const int* __restrict__ ei, const u64* __restrict__ T, u32 mask,
                       float* __restrict__ p_ksse, float* __restrict__ p_vsum,
                       u32* __restrict__ p_vcnt){
  __shared__ float shf[NT];
  __shared__ u32   shu[NT];
  const f32x4* kp4 = (const f32x4*)kp;
  const f32x4* kt4 = (const f32x4*)kt;
  const i32x4* i4  = (const i32x4*)ei;
  const i32x4* j4  = (const i32x4*)(ei + NE);
  float ksse = 0.f, vsum = 0.f; u32 vcnt = 0u;
  const u32 nq = NE/4;
  const u32 stride = NB_EDGE * NT;
  for (u32 q = blockIdx.x*NT + threadIdx.x; q < nq; q += stride){
    // streamed-once data: non-temporal, keep L2 for hash + kp gathers
    f32x4 a = __builtin_nontemporal_load(kp4 + q);
    f32x4 b = __builtin_nontemporal_load(kt4 + q);
    i32x4 iv = __builtin_nontemporal_load(i4 + q);
    i32x4 jv = __builtin_nontemporal_load(j4 + q);
    #define EDGE_ONE(av, bv, ic, jc) do {                               \
      float d_ = (av) - (bv); ksse += d_*d_;                            \
      u32 rk_ = (u32)(jc)*32768u + (u32)(ic);                           \
      int ri_ = hfind(T, rk_, mask);                                    \
      if (ri_ >= 0){ float v_ = (av) + kp[ri_]; vsum += v_*v_; vcnt++; } \
    } while(0)
    EDGE_ONE(a.x, b.x, iv.x, jv.x);
    EDGE_ONE(a.y, b.y, iv.y, jv.y);
    EDGE_ONE(a.z, b.z, iv.z, jv.z);
    EDGE_ONE(a.w, b.w, iv.w, jv.w);
    #undef EDGE_ONE
  }
  float rk = block_reduce_f(ksse, shf);
  float rv = block_reduce_f(vsum, shf);
  u32   rc = block_reduce_u(vcnt, shu);
  if (threadIdx.x == 0){
    p_ksse[blockIdx.x] = rk;
    p_vsum[blockIdx.x] = rv;
    p_vcnt[blockIdx.x] = rc;
  }
}

// ------------------------- node-level losses: occ / hybrid / bounds / energy
__global__ void k_node(const float* __restrict__ op, const float* __restrict__ ot,
                       const float* __restrict__ sp, const float* __restrict__ pp,
                       const float* __restrict__ dp, const float* __restrict__ fp,
                       const float* __restrict__ st, const float* __restrict__ pt,
                       const float* __restrict__ dt, const float* __restrict__ ft,
                       const float* __restrict__ ep, const float* __restrict__ et,
                       float* __restrict__ p_node){
  __shared__ float shf[NT];
  float occ = 0.f, hyb = 0.f, blo = 0.f, bhi = 0.f, en = 0.f;
  const u32 stride = NB_NODE * NT;
  const f32x4 *op4=(const f32x4*)op, *ot4=(const f32x4*)ot;
  const f32x4 *sp4=(const f32x4*)sp, *st4=(const f32x4*)st;
  const f32x4 *pp4=(const f32x4*)pp, *pt4=(const f32x4*)pt;
  const f32x4 *dp4=(const f32x4*)dp, *dt4=(const f32x4*)dt;
  const f32x4 *fp4=(const f32x4*)fp, *ft4=(const f32x4*)ft;
  for (u32 q = blockIdx.x*NT + threadIdx.x; q < NN/4; q += stride){
    f32x4 a = op4[q], b = ot4[q];   // occ_pred is re-read by k_cons: temporal
    occ += sq4(a, b);
    blo += relu2(-a.x) + relu2(-a.y) + relu2(-a.z) + relu2(-a.w);
    bhi += relu2(a.x-2.f) + relu2(a.y-2.f) + relu2(a.z-2.f) + relu2(a.w-2.f);
    hyb += sq4(sp4[q], st4[q]) + sq4(pp4[q], pt4[q])
         + sq4(dp4[q], dt4[q]) + sq4(fp4[q], ft4[q]);
  }
  const f32x4 *ep4=(const f32x4*)ep, *et4=(const f32x4*)et;
  for (u32 q = blockIdx.x*NT + threadIdx.x; q < NMOL/4; q += stride)
    en += sq4(ep4[q], et4[q]);
  float r0 = block_reduce_f(occ, shf);
  float r1 = block_reduce_f(hyb, shf);
  float r2 = block_reduce_f(blo, shf);
  float r3 = block_reduce_f(bhi, shf);
  float r4 = block_reduce_f(en,  shf);
  if (threadIdx.x == 0){
    p_node[blockIdx.x*5 + 0] = r0;
    p_node[blockIdx.x*5 + 1] = r1;
    p_node[blockIdx.x*5 + 2] = r2;
    p_node[blockIdx.x*5 + 3] = r3;
    p_node[blockIdx.x*5 + 4] = r4;
  }
}

// -------------------- conservation loss: per-molecule row sums via WMMA f32
// Each wave handles 16 molecules (32 nodes each). A[m,k] = partial sum of 8
// diffs; B = all-ones makes D[m,*] = molecule sum, exactly in f32 (RNE).
__global__ void k_cons(const float* __restrict__ op, const float* __restrict__ ot,
                       float* __restrict__ p_cons){
  const int lane  = threadIdx.x & 31;
  const int wave  = threadIdx.x >> 5;
  const int wglob = blockIdx.x * (NT/32) + wave;
  const int mol   = wglob * 16 + (lane & 15);
  const int base  = mol * 32 + ((lane & 16) ? 16 : 0);
  float ax = 0.f, ay = 0.f;
  #pragma unroll
  for (int t = 0; t < 8; ++t){
    ax += op[base + t]     - ot[base + t];
    ay += op[base + 8 + t] - ot[base + 8 + t];
  }
  float wave_sum;
#if defined(__gfx1250__) && __has_builtin(__builtin_amdgcn_wmma_f32_16x16x4_f32)
  v2f a; a.x = ax;  a.y = ay;     // A 16x4 f32: lanes 0-15 hold K=0,1; 16-31 hold K=2,3
  v2f b; b.x = 1.f; b.y = 1.f;    // B = ones -> layout-independent row sums
  v8f c = {};
  c = __builtin_amdgcn_wmma_f32_16x16x4_f32(
      /*neg_a=*/false, a, /*neg_b=*/false, b,
      /*c_mod=*/(short)0, c, /*reuse_a=*/false, /*reuse_b=*/false);
  // D layout: lane l, VGPR r -> M = r + 8*(l>=16); every N column identical.
  float s = 0.f;
  #pragma unroll
  for (int r = 0; r < 8; ++r) s += c[r] * c[r];
  wave_sum = __shfl(s, 0) + __shfl(s, 16);   // sum_m molsum[m]^2, m=0..15
#else
  float h  = ax + ay;
  float ms = h + __shfl_xor(h, 16);          // full 32-node molecule sum
  float sq = (lane < 16) ? ms * ms : 0.f;
  #pragma unroll
  for (int off = 8; off > 0; off >>= 1) sq += __shfl_down(sq, off);
  wave_sum = __shfl(sq, 0);
#endif
  __shared__ float wsh[NT/32];
  if (lane == 0) wsh[wave] = wave_sum;
  __syncthreads();
  if (threadIdx.x == 0){
    float t = 0.f;
    #pragma unroll
    for (int i = 0; i < NT/32; ++i) t += wsh[i];
    p_cons[blockIdx.x] = t;
  }
}

// ---------------------------------------------------------- final combine
__global__ void k_final(const float* __restrict__ p_ksse, const float* __restrict__ p_vsum,
                        const u32* __restrict__ p_vcnt, const float* __restrict__ p_node,
                        const float* __restrict__ p_cons,
                        const float* __restrict__ lvo_p, const float* __restrict__ lvk_p,
                        const float* __restrict__ lve_p, const float* __restrict__ lvh_p,
                        float* __restrict__ out){
  __shared__ float shf[NT];
  __shared__ u32   shu[NT];
  const int t = threadIdx.x;
  float ksse=0.f, vsum=0.f, cons=0.f, occ=0.f, hyb=0.f, blo=0.f, bhi=0.f, en=0.f;
  u32 vcnt = 0u;
  for (int i = t; i < NB_EDGE; i += NT){ ksse += p_ksse[i]; vsum += p_vsum[i]; vcnt += p_vcnt[i]; }
  for (int i = t; i < NB_NODE; i += NT){
    occ += p_node[i*5+0]; hyb += p_node[i*5+1];
    blo += p_node[i*5+2]; bhi += p_node[i*5+3]; en += p_node[i*5+4];
  }
  for (int i = t; i < NB_CONS; i += NT) cons += p_cons[i];
  ksse = block_reduce_f(ksse, shf);
  vsum = block_reduce_f(vsum, shf);
  cons = block_reduce_f(cons, shf);
  occ  = block_reduce_f(occ,  shf);
  hyb  = block_reduce_f(hyb,  shf);
  blo  = block_reduce_f(blo,  shf);
  bhi  = block_reduce_f(bhi,  shf);
  en   = block_reduce_f(en,   shf);
  vcnt = block_reduce_u(vcnt, shu);
  if (t == 0){
    const float lvo = lvo_p[0], lvk = lvk_p[0], lve = lve_p[0], lvh = lvh_p[0];
    float occ_l = occ / (float)NN;
    float kbo_l = ksse / (float)NE;
    float en_l  = en / (float)NMOL;
    float hyb_l = hyb / (float)(NN * 4);
    float anti  = (vcnt > 0u) ? (vsum / (float)vcnt) : 0.f;
    float cons_l = cons / (float)NMOL;
    float bnd   = 0.5f * (blo / (float)NN + bhi / (float)NN);
    float total = __expf(-lvo) * occ_l + lvo
                + __expf(-lvk) * kbo_l + lvk
                + __expf(-lve) * en_l  + lve
                + __expf(-lvh) * hyb_l + lvh
                + 0.1f * anti + 0.05f * cons_l + 0.01f * bnd;
    out[0] = total;
  }
}

extern "C" void kernel_launch(void* const* d_in, const int* in_sizes, int n_in,
                              void* d_out, int out_size, void* d_ws, size_t ws_size,
                              hipStream_t stream){
  const float* occ_p = (const float*)d_in[0];
  const float* kbo_p = (const float*)d_in[1];
  const float* en_p  = (const float*)d_in[2];
  const float* sp    = (const float*)d_in[3];
  const float* pp    = (const float*)d_in[4];
  const float* dp    = (const float*)d_in[5];
  const float* fp    = (const float*)d_in[6];
  const float* occ_t = (const float*)d_in[7];
  const float* kbo_t = (const float*)d_in[8];
  const float* en_t  = (const float*)d_in[9];
  const float* st    = (const float*)d_in[10];
  const float* pt    = (const float*)d_in[11];
  const float* dt    = (const float*)d_in[12];
  const float* ft    = (const float*)d_in[13];
  const float* lvo   = (const float*)d_in[14];
  const float* lvk   = (const float*)d_in[15];
  const float* lve   = (const float*)d_in[16];
  const float* lvh   = (const float*)d_in[17];
  const int*   ei    = (const int*)d_in[18];
  (void)in_sizes; (void)n_in; (void)out_size;  // batch (d_in[19]) is arange//32

  // Hash capacity: target 2^25 slots (0.5 load factor, 256 MiB -> mostly L2
  // resident), shrink deterministically if the workspace is smaller.
  u32 cap = 1u << 25;
  while ((u64)cap * 8ull + 65536ull > (u64)ws_size && cap > (1u << 16)) cap >>= 1;
  const u32 mask = cap - 1u;

  u64*   T      = (u64*)d_ws;
  float* pbase  = (float*)((char*)d_ws + (size_t)cap * 8u);
  float* p_ksse = pbase;
  float* p_vsum = pbase + NB_EDGE;
  u32*   p_vcnt = (u32*)(pbase + 2 * NB_EDGE);
  float* p_node = pbase + 3 * NB_EDGE;
  float* p_cons = p_node + NB_NODE * 5;

  k_init_hash<<<2048, NT, 0, stream>>>(T, cap);
  k_build_hash<<<2048, NT, 0, stream>>>(ei, T, mask);
  k_edge<<<NB_EDGE, NT, 0, stream>>>(kbo_p, kbo_t, ei, T, mask, p_ksse, p_vsum, p_vcnt);
  k_node<<<NB_NODE, NT, 0, stream>>>(occ_p, occ_t, sp, pp, dp, fp, st, pt, dt, ft,
                                     en_p, en_t, p_node);
  k_cons<<<NB_CONS, NT, 0, stream>>>(occ_p, occ_t, p_cons);
  k_final<<<1, NT, 0, stream>>>(p_ksse, p_vsum, p_vcnt, p_node, p_cons,
                                lvo, lvk, lve, lvh, (float*)d_out);
}